// TokenMixingAttn_59253368815799
// MI455X (gfx1250) — compile-verified
//
#include <hip/hip_runtime.h>
#include <hip/hip_bf16.h>
#include <math.h>
#include <stdint.h>

// Problem constants (from reference): B=8, T=512, C=32, D=128, P=1024
#define B_ 8
#define T_ 512
#define C_ 32
#define D_ 128
#define P_ 1024
#define EPS_ 1e-5f

static constexpr int TC_   = T_ * C_;      // 16384
static constexpr int NTOK_ = B_ * TC_;     // 131072 tokens (b,t,c)
static constexpr int LDSPITCH = 132;       // 132 floats = 528 B: 16B-aligned rows

typedef float v2f __attribute__((ext_vector_type(2)));
typedef float v8f __attribute__((ext_vector_type(8)));

// fp32 WMMA: D(16x16,f32) = A(16x4,f32) * B(4x16,f32) + C
// A layout: lane L: m=L&15, half h=L>>4; a.x=A[m][2h], a.y=A[m][2h+1] (per K-slice of 4)
// B layout: lane L: n=L&15, half h;      b.x=B[2h][n], b.y=B[2h+1][n]
// C layout: lane L: n=L&15; VGPR g holds row m = g + 8*(L>>4)
__device__ __forceinline__ v8f wmma_f32(v2f a, v2f b, v8f c) {
    return __builtin_amdgcn_wmma_f32_16x16x4_f32(false, a, false, b, (short)0, c, false, false);
}

// CDNA5 async global->LDS DMA (ASYNCcnt-tracked, no VGPR round trip).
// Generic pointers to __shared__ carry the DS offset in their low 32 bits
// (ISA §10.2).
__device__ __forceinline__ void async_b128_to_lds(void* lds_ptr, const void* gptr) {
    unsigned lds_off = (unsigned)(uintptr_t)lds_ptr;
    asm volatile("global_load_async_to_lds_b128 %0, %1, off"
                 :: "v"(lds_off), "v"(gptr) : "memory");
}
#define ASYNC_WAIT(N) asm volatile("s_wait_asynccnt %0" :: "n"(N) : "memory")

// Stage a 16 x 128 fp32 tile (contiguous rows of D_ floats) into padded LDS.
// One wave: 512 16-byte chunks -> 16 async B128 per lane.
__device__ __forceinline__ void stage_tile_async(float (*dst)[LDSPITCH],
                                                 const float* __restrict__ src) {
    int lane = threadIdx.x & 31;
#pragma unroll
    for (int j = 0; j < 16; ++j) {
        int idx = lane + j * 32;          // 0..511
        int r   = idx >> 5;               // row 0..15
        int ch  = idx & 31;               // 16B chunk within row
        async_b128_to_lds((char*)&dst[r][0] + ch * 16,
                          (const void*)(src + (size_t)r * D_ + ch * 4));
    }
}

// ---------------------------------------------------------------------------
// Kernel 1: channel gather + QKV projection (fp32 WMMA) + bias + RoPE rotate.
// Grid: NTOK_/16 blocks of 768 threads (24 waves).
// Wave w: mat = w/8 (0=q,1=k,2=v), colTile = w%8 (16 output dims each).
// Outputs written to workspace in (B*C, T, D) "head-major" layout.
// ---------------------------------------------------------------------------
__global__ __launch_bounds__(768)
void tok_qkv_rope_kernel(const float* __restrict__ x,
                         const unsigned char* __restrict__ x_mask,
                         const int* __restrict__ pos,
                         const float* __restrict__ pe,
                         const int* __restrict__ imp,
                         const int* __restrict__ idx_c,
                         const float* __restrict__ wq, const float* __restrict__ bq,
                         const float* __restrict__ wk, const float* __restrict__ bk,
                         const float* __restrict__ wv, const float* __restrict__ bv,
                         float* __restrict__ qo, float* __restrict__ ko,
                         float* __restrict__ vo)
{
    __shared__ __align__(16) float xg[16][LDSPITCH];  // gathered 16 x 128 tile
    __shared__ unsigned srcOff[16];      // source row base (element index into x)
    __shared__ unsigned outOff[16];      // dest row base ((b*C+c)*T+t)*D
    __shared__ int      posRow[16];

    const int tid     = threadIdx.x;
    const int rowBase = blockIdx.x * 16;

    if (tid < 16) {
        int rt  = rowBase + tid;
        int bb  = rt / TC_;
        int rem = rt - bb * TC_;
        int t   = rem / C_;
        int c   = rem - t * C_;
        int mo  = (bb * T_ + t) * C_ + c;
        int sel = x_mask[mo] ? imp[bb * T_ + t] : idx_c[mo];
        srcOff[tid] = (unsigned)(((bb * T_ + t) * C_ + sel) * D_);
        outOff[tid] = (unsigned)(((bb * C_ + c) * T_ + t) * D_);
        posRow[tid] = pos[mo];
    }
    __syncthreads();

    // async DMA gather of the 16x128 tile: 512 chunks of 16 B
    if (tid < 512) {
        int r  = tid >> 5;
        int ch = tid & 31;
        async_b128_to_lds((char*)&xg[r][0] + ch * 16,
                          (const void*)(x + (size_t)srcOff[r] + ch * 4));
    }
    ASYNC_WAIT(0);
    __syncthreads();

    const int wave    = tid >> 5;        // 0..23
    const int lane    = tid & 31;
    const int mat     = wave >> 3;       // 0=q 1=k 2=v
    const int colTile = wave & 7;
    const int n       = lane & 15;
    const int h       = lane >> 4;
    const int colBase = colTile * 16;

    const float* W  = (mat == 0) ? wq : (mat == 1) ? wk : wv;
    const float* Bi = (mat == 0) ? bq : (mat == 1) ? bk : bv;
    float*       O  = (mat == 0) ? qo : (mat == 1) ? ko : vo;

    // out[m][n] = sum_d xg[m][d] * W[n][d]  (torch Linear: x @ W.T)
    const v2f* arow = (const v2f*)&xg[n][0];                        // A rows (m == n lane slot)
    const v2f* wrow = (const v2f*)(W + (size_t)(colBase + n) * D_); // B[k][n] = W[n][k]

    v8f acc = {};
#pragma unroll
    for (int s = 0; s < 32; ++s) {
        v2f a = arow[2 * s + h];
        v2f b = wrow[2 * s + h];
        acc = wmma_f32(a, b, acc);
    }

    const float bias = Bi[colBase + n];
#pragma unroll
    for (int g = 0; g < 8; ++g) acc[g] += bias;

    if (mat < 2) {
        // RoPE: pair (even d -> re, odd d -> im); partner value is in lane^1.
        const int   j   = (colBase >> 1) + (n >> 1);      // pair index in [0,64)
        const float sgn = (n & 1) ? 1.0f : -1.0f;         // even: c*r0 - p*r1 ; odd: c*r0 + p*r1
#pragma unroll
        for (int g = 0; g < 8; ++g) {
            int   m  = g + 8 * h;
            int   p  = posRow[m];
            float r0 = pe[(size_t)p * D_ + 2 * j];
            float r1 = pe[(size_t)p * D_ + 2 * j + 1];
            float pr = __shfl_xor(acc[g], 1, 32);
            acc[g] = acc[g] * r0 + sgn * pr * r1;
        }
    }

#pragma unroll
    for (int g = 0; g < 8; ++g) {
        int m = g + 8 * h;
        O[(size_t)outOff[m] + colBase + n] = acc[g];
    }
}

// ---------------------------------------------------------------------------
// Kernel 2: per-head flash attention + key-weight + residual + LS + LayerNorm.
// One wave (32 threads) per (head, q-tile-of-16). Grid = B*C * (T/16) = 8192.
// Sᵀ = K_tile · Qᵀ so the softmax axis (k) lies on VGPRs + one half-shuffle.
// V tiles are double-buffered with async global->LDS DMA.
// ---------------------------------------------------------------------------
__global__ __launch_bounds__(32)
void tok_attn_ln_kernel(const float* __restrict__ q,
                        const float* __restrict__ k,
                        const float* __restrict__ v,
                        const unsigned char* __restrict__ x_mask,
                        const unsigned char* __restrict__ p_mask,
                        const float* __restrict__ x,
                        const float* __restrict__ ls_g,
                        const float* __restrict__ ln_g,
                        const float* __restrict__ ln_b,
                        float* __restrict__ out)
{
    __shared__ __align__(16) float qL[16][LDSPITCH];     // Q tile (unscaled)
    __shared__ __align__(16) float vL[2][16][LDSPITCH];  // double-buffered V tiles
    __shared__ float pL[16][18];                          // P (q x k), weighted probs

    const int lane = threadIdx.x;
    const int n    = lane & 15;
    const int h    = lane >> 4;
    const int head = blockIdx.x >> 5;   // b*C + c
    const int qt   = blockIdx.x & 31;
    const int bb   = head >> 5;
    const int c    = head & 31;

    const float scale = 0.08838834764831845f;   // 1/sqrt(128), applied to scores

    const float* headQ = q + (size_t)head * T_ * D_;
    const float* headK = k + (size_t)head * T_ * D_;
    const float* headV = v + (size_t)head * T_ * D_;

    // async-stage Q tile and V tile 0 (in-order completion on ASYNCcnt)
    stage_tile_async(qL, headQ + (size_t)(qt * 16) * D_);
    stage_tile_async(vL[0], headV);

    const bool pmq = p_mask[((size_t)bb * T_ + qt * 16 + n) * C_ + c] != 0;

    float mrun = -INFINITY, lrun = 0.0f;
    v8f o[8];
#pragma unroll
    for (int dt = 0; dt < 8; ++dt) o[dt] = (v8f){};

    for (int kt = 0; kt < 32; ++kt) {
        // prefetch next V tile into the other buffer, then wait for current
        if (kt + 1 < 32) {
            stage_tile_async(vL[(kt + 1) & 1], headV + (size_t)((kt + 1) * 16) * D_);
            ASYNC_WAIT(16);     // Q + V[kt] complete; V[kt+1] may still be in flight
        } else {
            ASYNC_WAIT(0);
        }
        __syncthreads();

        const size_t kBase = (size_t)(kt * 16) * D_;
        const float (*vT)[LDSPITCH] = vL[kt & 1];

        // S^T[k][q] = sum_d K[k][d] * Q[q][d]
        const v2f* arow = (const v2f*)(headK + kBase + (size_t)n * D_);
        const v2f* brow = (const v2f*)&qL[n][0];
        v8f sacc = {};
#pragma unroll
        for (int s = 0; s < 32; ++s) {
            v2f a = arow[2 * s + h];
            v2f b = brow[2 * s + h];
            sacc = wmma_f32(a, b, sacc);
        }

        // masked online softmax along k (VGPR axis + half shuffle)
        float sv[8];
        float tmax = -INFINITY;
#pragma unroll
        for (int g = 0; g < 8; ++g) {
            int  tk     = kt * 16 + g + 8 * h;
            bool pmk    = p_mask[((size_t)bb * T_ + tk) * C_ + c] != 0;
            bool masked = pmk || pmq;
            float s     = masked ? -INFINITY : sacc[g] * scale;
            sv[g]   = s;
            tmax    = fmaxf(tmax, s);
        }
        tmax = fmaxf(tmax, __shfl_xor(tmax, 16, 32));
        float mnew  = fmaxf(mrun, tmax);
        float alpha = (mnew == -INFINITY) ? 0.0f : expf(mrun - mnew);

        float psum = 0.0f;
#pragma unroll
        for (int g = 0; g < 8; ++g) {
            float pv = (sv[g] == -INFINITY) ? 0.0f : expf(sv[g] - mnew);
            psum += pv;
            int   tk = kt * 16 + g + 8 * h;
            // per-key weight applied AFTER normalization -> fold into P, not into lrun
            float kw = x_mask[((size_t)bb * T_ + tk) * C_ + c] ? (1.0f / (float)T_) : 1.0f;
            pL[n][g + 8 * h] = pv * kw;
        }
        psum += __shfl_xor(psum, 16, 32);
        lrun = lrun * alpha + psum;
        mrun = mnew;
        __syncthreads();

        // broadcast alpha from softmax lanes (q = lane&15) to out rows (q on VGPR axis)
        float am[8];
#pragma unroll
        for (int g = 0; g < 8; ++g) am[g] = __shfl(alpha, g + 8 * h, 32);

        const v2f* prow = (const v2f*)&pL[n][0];
#pragma unroll
        for (int dt = 0; dt < 8; ++dt) {
            v8f oc = o[dt];
#pragma unroll
            for (int g = 0; g < 8; ++g) oc[g] *= am[g];
#pragma unroll
            for (int s = 0; s < 4; ++s) {
                v2f a = prow[2 * s + h];
                v2f b;
                b.x = vT[4 * s + 2 * h][dt * 16 + n];
                b.y = vT[4 * s + 2 * h + 1][dt * 16 + n];
                oc = wmma_f32(a, b, oc);
            }
            o[dt] = oc;
        }
        __syncthreads();
    }

    // 1/l (0 for fully-masked rows -> matches nan_to_num semantics)
    float rl = (lrun > 0.0f) ? (1.0f / lrun) : 0.0f;
    float rm[8];
#pragma unroll
    for (int g = 0; g < 8; ++g) rm[g] = __shfl(rl, g + 8 * h, 32);

    // y = x + ls_gamma * attn_out  (in-register, reusing o[])
#pragma unroll
    for (int dt = 0; dt < 8; ++dt) {
        int   d  = dt * 16 + n;
        float ls = ls_g[d];
#pragma unroll
        for (int g = 0; g < 8; ++g) {
            int    t  = qt * 16 + g + 8 * h;
            size_t xo = (((size_t)bb * T_ + t) * C_ + c) * (size_t)D_ + d;
            o[dt][g] = x[xo] + ls * (o[dt][g] * rm[g]);
        }
    }

    // LayerNorm over d=128: per row (VGPR g, half h) reduce across 16 lanes of half
    float mu[8], rstd[8];
#pragma unroll
    for (int g = 0; g < 8; ++g) {
        float s = 0.0f;
#pragma unroll
        for (int dt = 0; dt < 8; ++dt) s += o[dt][g];
#pragma unroll
        for (int off = 1; off < 16; off <<= 1) s += __shfl_xor(s, off, 32);
        mu[g] = s * (1.0f / (float)D_);
    }
#pragma unroll
    for (int g = 0; g < 8; ++g) {
        float s = 0.0f;
#pragma unroll
        for (int dt = 0; dt < 8; ++dt) {
            float d0 = o[dt][g] - mu[g];
            s += d0 * d0;
        }
#pragma unroll
        for (int off = 1; off < 16; off <<= 1) s += __shfl_xor(s, off, 32);
        rstd[g] = rsqrtf(s * (1.0f / (float)D_) + EPS_);
    }

#pragma unroll
    for (int dt = 0; dt < 8; ++dt) {
        int   d  = dt * 16 + n;
        float gg = ln_g[d];
        float be = ln_b[d];
#pragma unroll
        for (int g = 0; g < 8; ++g) {
            int    t  = qt * 16 + g + 8 * h;
            size_t xo = (((size_t)bb * T_ + t) * C_ + c) * (size_t)D_ + d;
            out[xo] = (o[dt][g] - mu[g]) * rstd[g] * gg + be;
        }
    }
}

// ---------------------------------------------------------------------------
extern "C" void kernel_launch(void* const* d_in, const int* in_sizes, int n_in,
                              void* d_out, int out_size, void* d_ws, size_t ws_size,
                              hipStream_t stream) {
    (void)in_sizes; (void)n_in; (void)out_size;

    const float*         x      = (const float*)d_in[0];
    const unsigned char* x_mask = (const unsigned char*)d_in[1];
    const unsigned char* p_mask = (const unsigned char*)d_in[2];
    const int*           pos    = (const int*)d_in[3];
    const float*         pe     = (const float*)d_in[4];
    const int*           imp    = (const int*)d_in[5];
    const int*           idx_c  = (const int*)d_in[6];
    const float*         wq     = (const float*)d_in[7];
    const float*         bq     = (const float*)d_in[8];
    const float*         wk     = (const float*)d_in[9];
    const float*         bk     = (const float*)d_in[10];
    const float*         wv     = (const float*)d_in[11];
    const float*         bv     = (const float*)d_in[12];
    const float*         lsg    = (const float*)d_in[13];
    const float*         lng    = (const float*)d_in[14];
    const float*         lnb    = (const float*)d_in[15];
    float*               out    = (float*)d_out;

    // q/k/v workspace in (B*C, T, D) layout: 3 * 64 MB = 192 MB
    const size_t per = (size_t)B_ * C_ * T_ * D_;
    float* qws = (float*)d_ws;
    float* kws = qws + per;
    float* vws = kws + per;
    (void)ws_size;

    tok_qkv_rope_kernel<<<dim3(NTOK_ / 16), dim3(768), 0, stream>>>(
        x, x_mask, pos, pe, imp, idx_c,
        wq, bq, wk, bk, wv, bv,
        qws, kws, vws);

    tok_attn_ln_kernel<<<dim3(B_ * C_ * (T_ / 16)), dim3(32), 0, stream>>>(
        qws, kws, vws, x_mask, p_mask, x, lsg, lng, lnb, out);
}